// EighMSE_2_42880953483665
// MI455X (gfx1250) — compile-verified
//
#include <hip/hip_runtime.h>
#include <hip/hip_bf16.h>

typedef __attribute__((ext_vector_type(2))) float v2f;
typedef __attribute__((ext_vector_type(8))) float v8f;

// ---------------------------------------------------------------------------
// Exact f32 wave(32) sum using V_WMMA_F32_16X16X4_F32.
// A (16x4 f32): VGPR0 lanes0-15 -> (M=lane,K=0), lanes16-31 -> (M=lane-16,K=2);
// VGPR1 = K=1/K=3 (set to 0).  B = ones(4x16).
// => D[m,n] = x[m] + x[m+16]  (any n).
// Lanes 0-15 sum d[0..7] = sum(x[0..7]) + sum(x[16..23]);
// lanes 16-31 get sum(x[8..15]) + sum(x[24..31]).  One shfl_xor(16) completes.
// Requires EXEC == all ones (call from fully converged code only).
// ---------------------------------------------------------------------------
__device__ __forceinline__ float wave_sum_wmma(float x) {
    v2f a; a[0] = x;    a[1] = 0.0f;
    v2f b; b[0] = 1.0f; b[1] = 1.0f;
    v8f c = {};
    v8f d = __builtin_amdgcn_wmma_f32_16x16x4_f32(
        /*neg_a=*/false, a, /*neg_b=*/false, b,
        /*c_mod=*/(short)0, c, /*reuse_a=*/false, /*reuse_b=*/false);
    float s = ((d[0] + d[1]) + (d[2] + d[3])) + ((d[4] + d[5]) + (d[6] + d[7]));
    s += __shfl_xor(s, 16);
    return s;  // every lane holds the full 32-lane sum
}

// ---------------------------------------------------------------------------
// Closed-form 2x2 symmetric eig, mirroring the reference literally.
// ---------------------------------------------------------------------------
struct Eig2 {
    float e1, e2;      // analytic evals: larger, smaller
    float v1x, v1y;    // eigenvector matched to e1
    float v2x, v2y;    // eigenvector matched to e2
};

__device__ __forceinline__ Eig2 eig2(float a, float b, float d) {
    float s    = a + d;
    // _analytic_evals: disc = sqrt((-a-d)^2 - 4(ad - b^2))
    float disc = sqrtf(s * s - 4.0f * (a * d - b * b));
    float e1   = (s + disc) * 0.5f;   // larger
    float e2   = (s - disc) * 0.5f;   // smaller

    // _sym2x2_eigh
    float mean   = s * 0.5f;
    float hd     = (a - d) * 0.5f;
    float r      = sqrtf(hd * hd + b * b);
    float lam_lo = mean - r;
    float lam_hi = mean + r;
    float vx = b;
    float vy = lam_hi - a;
    float n  = sqrtf(vx * vx + vy * vy);
    bool  safe = n > 1e-20f;
    float inv  = safe ? (1.0f / n) : 0.0f;
    float vhx  = safe ? vx * inv : 1.0f;
    float vhy  = safe ? vy * inv : 0.0f;
    // evecs columns: col0 = v_lo = (-vhy, vhx), col1 = v_hi = (vhx, vhy)
    // evals order:   [lam_lo, lam_hi]
    // argmin (first index on tie -> use <=)
    bool c1 = fabsf(lam_lo - e1) <= fabsf(lam_hi - e1);
    bool c2 = fabsf(lam_lo - e2) <= fabsf(lam_hi - e2);

    Eig2 o;
    o.e1 = e1;  o.e2 = e2;
    o.v1x = c1 ? -vhy : vhx;   o.v1y = c1 ? vhx : vhy;
    o.v2x = c2 ? -vhy : vhx;   o.v2y = c2 ? vhx : vhy;
    return o;
}

struct Acc5 { float s0, s1, s2, s3, s4; };

__device__ __forceinline__ void accum_one(Acc5& acc,
                                          float pa, float pb, float pd,
                                          float ta, float tb, float td) {
    Eig2 p = eig2(pa, pb, pd);
    Eig2 t = eig2(ta, tb, td);
    float de1 = p.e1 - t.e1;
    float de2 = p.e2 - t.e2;
    acc.s0 += de1 * de1 + de2 * de2;          // evals: mean over B*2
    float d10 = p.v1x - t.v1x; acc.s1 += d10 * d10;
    float d11 = p.v1y - t.v1y; acc.s2 += d11 * d11;
    float d20 = p.v2x - t.v2x; acc.s3 += d20 * d20;
    float d21 = p.v2y - t.v2y; acc.s4 += d21 * d21;
}

// ---------------------------------------------------------------------------
// Main pass: grid-stride map over groups of 4 elements (3x b128 loads per
// array per group), hierarchical deterministic reduction.
// partials[block*5 + k] = block's sum of component k (no atomics).
// ---------------------------------------------------------------------------
__global__ void eigh_mse_main(const float* __restrict__ yp,
                              const float* __restrict__ yt,
                              int B, float* __restrict__ partials) {
    const int tid    = blockIdx.x * blockDim.x + threadIdx.x;
    const int stride = gridDim.x * blockDim.x;
    const int B4     = B >> 2;           // number of 4-element groups

    const float4* __restrict__ yp4 = (const float4*)yp;
    const float4* __restrict__ yt4 = (const float4*)yt;

    Acc5 acc = {0.0f, 0.0f, 0.0f, 0.0f, 0.0f};

    for (int g = tid; g < B4; g += stride) {
        size_t b4 = (size_t)g * 3;       // three float4 = 48B = 4 elements
        // prefetch next chunk (speculative, OOB-safe)
        __builtin_prefetch(&yp4[b4 + (size_t)stride * 3], 0, 0);
        __builtin_prefetch(&yt4[b4 + (size_t)stride * 3], 0, 0);

        float4 p0 = yp4[b4 + 0], p1 = yp4[b4 + 1], p2 = yp4[b4 + 2];
        float4 t0 = yt4[b4 + 0], t1 = yt4[b4 + 1], t2 = yt4[b4 + 2];

        // element layout within the 12 floats (row-major [4,3]):
        accum_one(acc, p0.x, p0.y, p0.z,  t0.x, t0.y, t0.z);
        accum_one(acc, p0.w, p1.x, p1.y,  t0.w, t1.x, t1.y);
        accum_one(acc, p1.z, p1.w, p2.x,  t1.z, t1.w, t2.x);
        accum_one(acc, p2.y, p2.z, p2.w,  t2.y, t2.z, t2.w);
    }

    // Scalar tail for B % 4 (handled by first few threads of block 0;
    // divergent here, but fully reconverged before the WMMA below).
    const int rem = B & 3;
    if (blockIdx.x == 0 && threadIdx.x < rem) {
        size_t base = (size_t)(B4 * 4 + threadIdx.x) * 3;
        accum_one(acc, yp[base], yp[base + 1], yp[base + 2],
                       yt[base], yt[base + 1], yt[base + 2]);
    }

    // All threads reconverged here: EXEC all ones -> WMMA wave reduction.
    float r0 = wave_sum_wmma(acc.s0);
    float r1 = wave_sum_wmma(acc.s1);
    float r2 = wave_sum_wmma(acc.s2);
    float r3 = wave_sum_wmma(acc.s3);
    float r4 = wave_sum_wmma(acc.s4);

    __shared__ float sblk[8][5];  // up to 8 waves per 256-thread block
    const int wave = threadIdx.x >> 5;
    const int lane = threadIdx.x & 31;
    if (lane == 0) {
        sblk[wave][0] = r0; sblk[wave][1] = r1; sblk[wave][2] = r2;
        sblk[wave][3] = r3; sblk[wave][4] = r4;
    }
    __syncthreads();
    if (threadIdx.x < 5) {
        const int nw = blockDim.x >> 5;
        float a = 0.0f;
        for (int w = 0; w < nw; ++w) a += sblk[w][threadIdx.x];
        partials[(size_t)blockIdx.x * 5 + threadIdx.x] = a;  // deterministic
    }
}

// ---------------------------------------------------------------------------
// Final combine: one wave. Deterministic column sums + weighted means.
// ---------------------------------------------------------------------------
__global__ void eigh_mse_final(const float* __restrict__ partials, int nblocks,
                               const float* __restrict__ weights,
                               float* __restrict__ out, float fB) {
    const int lane = threadIdx.x;  // 0..31
    float acc[5];
#pragma unroll
    for (int k = 0; k < 5; ++k) {
        float a = 0.0f;
        for (int j = lane; j < nblocks; j += 32)
            a += partials[(size_t)j * 5 + k];
        acc[k] = a;
    }
    float t0 = wave_sum_wmma(acc[0]);
    float t1 = wave_sum_wmma(acc[1]);
    float t2 = wave_sum_wmma(acc[2]);
    float t3 = wave_sum_wmma(acc[3]);
    float t4 = wave_sum_wmma(acc[4]);
    if (lane == 0) {
        float invB  = 1.0f / fB;
        float inv2B = 1.0f / (2.0f * fB);
        float mse = weights[0] * (t0 * inv2B)
                  + weights[1] * (t1 * invB)
                  + weights[2] * (t2 * invB)
                  + weights[3] * (t3 * invB)
                  + weights[4] * (t4 * invB);
        out[0] = mse;
    }
}

// ---------------------------------------------------------------------------
extern "C" void kernel_launch(void* const* d_in, const int* in_sizes, int n_in,
                              void* d_out, int out_size, void* d_ws, size_t ws_size,
                              hipStream_t stream) {
    const float* yp = (const float*)d_in[0];   // y_pred [B,3]
    const float* yt = (const float*)d_in[1];   // y_true [B,3]
    const float* w  = (const float*)d_in[2];   // weights [5]
    float* out      = (float*)d_out;

    const int B = in_sizes[0] / 3;

    const int threads = 256;
    int blocks = 1024;                                   // 16 elems/thread at B=4M
    int maxBlocksByWs = (int)(ws_size / (5 * sizeof(float)));
    if (blocks > maxBlocksByWs) blocks = maxBlocksByWs;
    int maxBlocksByB = ((B >> 2) + threads - 1) / threads;
    if (blocks > maxBlocksByB) blocks = maxBlocksByB;
    if (blocks < 1) blocks = 1;

    float* partials = (float*)d_ws;  // blocks*5 floats, fully overwritten

    eigh_mse_main<<<blocks, threads, 0, stream>>>(yp, yt, B, partials);
    eigh_mse_final<<<1, 32, 0, stream>>>(partials, blocks, w, out, (float)B);
}